// EmbedGraphConv_34153579937817
// MI455X (gfx1250) — compile-verified
//
#include <hip/hip_runtime.h>

// EmbedGraphConv on gfx1250 (MI455X):
//   h   = embedding[feat] * clip(outdeg,1)^-1/2
//   out = segment_sum(h[src], dst) * clip(indeg,1)^-1/2 + bias
// N=50000, E=800000, D=128, VOCAB=50000.  Embedding (25.6MB), output (25.6MB)
// and edge lists (~10MB) are all L2-resident (192MB L2); the op is pure
// gather/scatter bandwidth, AI ~0.25 FLOP/byte -> no matrix-core work exists.
//
// Pipeline: degree histogram -> hierarchical wave32 scan (CSR offsets) ->
// counting-sort edges by dst -> one wave32 per node gather-reduce with
// batched edge-header resolution (32 headers in parallel, shfl broadcast,
// 32 independent b128 row loads in flight) and fused norm+bias b128 store.
// No fp atomics anywhere; every output row written exactly once.

#define DFEAT 128
#define SCAN_TB 1024

// ---------------------------------------------------------------------------
// Kernel 0: zero degree counters.
// ---------------------------------------------------------------------------
__global__ void egc_zero_kernel(unsigned* __restrict__ out_cnt,
                                unsigned* __restrict__ in_cnt,
                                int n_nodes) {
    int i = blockIdx.x * blockDim.x + threadIdx.x;
    if (i < n_nodes) { out_cnt[i] = 0u; in_cnt[i] = 0u; }
}

// ---------------------------------------------------------------------------
// Kernel 1: degree histogram (u32 atomics, L2-resident).
// ---------------------------------------------------------------------------
__global__ void egc_degree_kernel(const int* __restrict__ src,
                                  const int* __restrict__ dst,
                                  unsigned* __restrict__ out_cnt,
                                  unsigned* __restrict__ in_cnt,
                                  int n_edges) {
    int e = blockIdx.x * blockDim.x + threadIdx.x;
    if (e < n_edges) {
        atomicAdd(&out_cnt[src[e]], 1u);
        atomicAdd(&in_cnt[dst[e]], 1u);
    }
}

// ---------------------------------------------------------------------------
// Kernel 2: single-workgroup exclusive scan of in-degrees -> CSR offsets.
// Hierarchical wave32 scan: 5-step shfl_up per wave (lockstep, no barriers),
// wave0 scans the 32 wave totals, 2 barriers per 1024-chunk.
// ---------------------------------------------------------------------------
__device__ __forceinline__ unsigned wave_incl_scan(unsigned x, int lane) {
    #pragma unroll
    for (int off = 1; off < 32; off <<= 1) {
        unsigned y = __shfl_up(x, off, 32);
        if (lane >= off) x += y;
    }
    return x;
}

__global__ void egc_scan_kernel(const unsigned* __restrict__ in_cnt,
                                unsigned* __restrict__ offsets,   // [n+1]
                                unsigned* __restrict__ cursor,    // [n]
                                int n) {
    __shared__ unsigned wave_tot[SCAN_TB / 32];
    __shared__ unsigned chunk_tot;
    const int tid  = threadIdx.x;
    const int lane = tid & 31;
    const int wid  = tid >> 5;
    unsigned running = 0u;

    for (int base = 0; base < n; base += SCAN_TB) {
        const int i = base + tid;
        const unsigned v = (i < n) ? in_cnt[i] : 0u;

        const unsigned incl = wave_incl_scan(v, lane);   // within-wave inclusive
        if (lane == 31) wave_tot[wid] = incl;
        __syncthreads();

        if (wid == 0) {
            const unsigned t  = wave_tot[lane];
            const unsigned tx = wave_incl_scan(t, lane); // scan wave totals
            wave_tot[lane] = tx - t;                     // exclusive per-wave base
            if (lane == 31) chunk_tot = tx;              // chunk grand total
        }
        __syncthreads();

        const unsigned excl = running + wave_tot[wid] + (incl - v);
        if (i < n) {
            offsets[i] = excl;
            cursor[i]  = excl;
        }
        running += chunk_tot;   // uniform; barrier ordering makes this safe
    }
    if (tid == 0) offsets[n] = running;
}

// ---------------------------------------------------------------------------
// Kernel 3: bucket edge ids by dst (counting-sort placement).
// ---------------------------------------------------------------------------
__global__ void egc_bucket_kernel(const int* __restrict__ dst,
                                  unsigned* __restrict__ cursor,
                                  unsigned* __restrict__ perm,
                                  int n_edges) {
    int e = blockIdx.x * blockDim.x + threadIdx.x;
    if (e < n_edges) {
        const unsigned pos = atomicAdd(&cursor[dst[e]], 1u);
        perm[pos] = (unsigned)e;
    }
}

// ---------------------------------------------------------------------------
// Kernel 4: gather-reduce, one wave32 per dst node.
// Phase A (headers): lane l resolves edge base+l -> (feat row, rsqrt(outdeg))
//   -- up to 32 pointer-chase chains run in parallel instead of serially --
//   and prefetches the first cacheline of its row (global_prefetch_b8).
// Phase B (sweep): broadcast (f, nl) from lane j via shfl; all 32 lanes load
//   one float4 of row f (global_load_b128) and FMA into the accumulator.
//   Loads across j are independent -> high memory-level parallelism.
// Epilogue: fused indeg^-1/2 + bias, single b128 store per lane.
// ---------------------------------------------------------------------------
__global__ void egc_gather_kernel(const int* __restrict__ feat,
                                  const int* __restrict__ src,
                                  const float* __restrict__ embedding,
                                  const unsigned* __restrict__ out_cnt,
                                  const unsigned* __restrict__ offsets,
                                  const unsigned* __restrict__ perm,
                                  const float* __restrict__ bias,
                                  float* __restrict__ out,
                                  int n_nodes) {
    const int lane = threadIdx.x & 31;
    const int node = blockIdx.x * (blockDim.x >> 5) + (threadIdx.x >> 5);
    if (node >= n_nodes) return;

    const unsigned beg = offsets[node];
    const unsigned end = offsets[node + 1];

    float4 acc = make_float4(0.f, 0.f, 0.f, 0.f);

    for (unsigned base = beg; base < end; base += 32u) {
        // ---- Phase A: parallel header resolution (one edge per lane) ----
        const unsigned k = base + (unsigned)lane;
        int   f  = 0;
        float nl = 0.0f;
        if (k < end) {
            const int e = (int)perm[k];
            const int s = src[e];
            f  = feat[s];
            nl = rsqrtf(fmaxf((float)out_cnt[s], 1.0f));
            __builtin_prefetch(embedding + (size_t)f * DFEAT, 0, 1);
        }
        const int cnt = (int)min(32u, end - base);

        // ---- Phase B: sweep the batch, independent b128 row loads ----
        for (int j = 0; j < cnt; ++j) {
            const int   fj  = __shfl(f, j, 32);
            const float nlj = __shfl(nl, j, 32);
            const float4 v =
                ((const float4*)(embedding + (size_t)fj * DFEAT))[lane];
            acc.x += v.x * nlj;
            acc.y += v.y * nlj;
            acc.z += v.z * nlj;
            acc.w += v.w * nlj;
        }
    }

    const float  nr = rsqrtf(fmaxf((float)(end - beg), 1.0f));
    const float4 bv = ((const float4*)bias)[lane];
    float4 r;
    r.x = acc.x * nr + bv.x;
    r.y = acc.y * nr + bv.y;
    r.z = acc.z * nr + bv.z;
    r.w = acc.w * nr + bv.w;
    ((float4*)(out + (size_t)node * DFEAT))[lane] = r;
}

// ---------------------------------------------------------------------------
extern "C" void kernel_launch(void* const* d_in, const int* in_sizes, int n_in,
                              void* d_out, int out_size, void* d_ws, size_t ws_size,
                              hipStream_t stream) {
    // setup_inputs order: feat[N], src[E], dst[E], embedding[VOCAB*D], bias[D]
    const int*   feat      = (const int*)d_in[0];
    const int*   src       = (const int*)d_in[1];
    const int*   dst       = (const int*)d_in[2];
    const float* embedding = (const float*)d_in[3];
    const float* bias      = (const float*)d_in[4];
    float*       out       = (float*)d_out;

    const int n_nodes = in_sizes[0];
    const int n_edges = in_sizes[1];

    // workspace (u32 each): out_cnt[N] | in_cnt[N] | offsets[N+1] |
    //                       cursor[N]  | perm[E]              (~4.0 MB)
    unsigned* out_cnt = (unsigned*)d_ws;
    unsigned* in_cnt  = out_cnt + n_nodes;
    unsigned* offsets = in_cnt  + n_nodes;
    unsigned* cursor  = offsets + (n_nodes + 1);
    unsigned* perm    = cursor  + n_nodes;

    const int TB = 256;

    // 0) zero counters
    egc_zero_kernel<<<(n_nodes + TB - 1) / TB, TB, 0, stream>>>(out_cnt, in_cnt,
                                                                n_nodes);
    // 1) degrees
    egc_degree_kernel<<<(n_edges + TB - 1) / TB, TB, 0, stream>>>(src, dst,
                                                                  out_cnt, in_cnt,
                                                                  n_edges);
    // 2) CSR offsets (single workgroup, hierarchical wave32 scan)
    egc_scan_kernel<<<1, SCAN_TB, 0, stream>>>(in_cnt, offsets, cursor, n_nodes);

    // 3) bucket edges by dst
    egc_bucket_kernel<<<(n_edges + TB - 1) / TB, TB, 0, stream>>>(dst, cursor,
                                                                  perm, n_edges);
    // 4) gather-reduce: one wave32 per node, 8 nodes per 256-thread block
    {
        const int nodes_per_block = TB / 32;
        const int blocks = (n_nodes + nodes_per_block - 1) / nodes_per_block;
        egc_gather_kernel<<<blocks, TB, 0, stream>>>(feat, src, embedding,
                                                     out_cnt, offsets, perm,
                                                     bias, out, n_nodes);
    }
}